// TemporalGNNSimple_9569187135762
// MI455X (gfx1250) — compile-verified
//
#include <hip/hip_runtime.h>
#include <hip/hip_bf16.h>
#include <math.h>

// Problem constants (match reference)
#define B_    4
#define T_    30
#define N_    2000
#define E_    2000
#define FIN   8
#define FHID  64
#define FGRU  128
#define G_    (B_ * T_)          // 120 frames
#define NCLS  2

typedef __attribute__((ext_vector_type(2))) float v2f;
typedef __attribute__((ext_vector_type(4))) float v4f;
typedef __attribute__((ext_vector_type(8))) float v8f;

// ---------------------------------------------------------------------------
// Degree count: deg[f][dst] += 1 per edge (self-loop "+1" folded into rsqrt).
// ---------------------------------------------------------------------------
__global__ void deg_kernel(const int* __restrict__ ei, float* __restrict__ deg) {
    int idx = blockIdx.x * blockDim.x + threadIdx.x;      // f*E + e
    if (idx >= G_ * E_) return;
    int f = idx / E_, e = idx - f * E_;
    int dst = ei[f * 2 * E_ + E_ + e];
    atomicAdd(&deg[f * N_ + dst], 1.0f);
}

// dis = rsqrt(deg + 1)   (self-loop guarantees deg_total >= 1, no zero case)
__global__ void rsqrt_kernel(float* __restrict__ d) {
    int idx = blockIdx.x * blockDim.x + threadIdx.x;
    if (idx < G_ * N_) d[idx] = rsqrtf(d[idx] + 1.0f);
}

// ---------------------------------------------------------------------------
// Per-frame GEMM with fp32 WMMA:  C_f[N_ x 64] = A_f[N_ x K] @ W[K x 64]
//
// One wave owns a 16(M) x 64(N) strip: 4 accumulators, so each 16x4 A
// fragment is loaded once and fed to 4 back-to-back V_WMMA_F32_16X16X4_F32.
// W (K x 64, <=16 KB) is staged in LDS cooperatively by the 4 waves of the
// block; B fragments then come from ds_load instead of repeated L2 hits.
//
// Fragment layouts (ISA 7.12.2, 32-bit):
//   A 16x4 : lane = M%16; lanes 0-15 hold K={0,1}, lanes 16-31 hold K={2,3}
//   B 4x16 : lane = N%16; same K-half split
//   C 16x16: 8 VGPRs; lanes 16-31 carry rows M+8
// ---------------------------------------------------------------------------
template <int K>
__global__ void gemm_wmma(const float* __restrict__ A,
                          const float* __restrict__ W,
                          float* __restrict__ C) {
    __shared__ float Wl[K * FHID];

    const int lane = threadIdx.x;                          // 0..31
    const int tid  = threadIdx.y * 32 + lane;              // 0..127

    // Cooperative stage of W into LDS (float4 granularity).
    const v4f* W4 = (const v4f*)W;
    v4f* Wl4 = (v4f*)Wl;
#pragma unroll
    for (int i = tid; i < (K * FHID) / 4; i += 128) Wl4[i] = W4[i];
    __syncthreads();

    const int MTILES = N_ / 16;                            // 125 per frame
    const int tile = blockIdx.x * 4 + threadIdx.y;         // wave-uniform
    if (tile >= G_ * MTILES) return;                       // uniform: EXEC stays all-1s
    const int f  = tile / MTILES;
    const int mt = tile - f * MTILES;

    const float* Af = A + (size_t)f * N_ * K + (size_t)(mt * 16) * K;
    const int row   = lane & 15;          // M within tile (A) / N within tile (B)
    const int khalf = lane >> 4;          // 0: K pair {0,1}, 1: K pair {2,3}

    v8f acc0 = {}, acc1 = {}, acc2 = {}, acc3 = {};
#pragma unroll
    for (int k0 = 0; k0 < K; k0 += 4) {
        const int ka = k0 + khalf * 2;
        v2f a;
        a.x = Af[row * K + ka];
        a.y = Af[row * K + ka + 1];
        const float* w0 = &Wl[(ka + 0) * FHID + row];
        const float* w1 = &Wl[(ka + 1) * FHID + row];
        v2f b0; b0.x = w0[0];  b0.y = w1[0];
        v2f b1; b1.x = w0[16]; b1.y = w1[16];
        v2f b2; b2.x = w0[32]; b2.y = w1[32];
        v2f b3; b3.x = w0[48]; b3.y = w1[48];
        acc0 = __builtin_amdgcn_wmma_f32_16x16x4_f32(false, a, false, b0, (short)0, acc0, false, false);
        acc1 = __builtin_amdgcn_wmma_f32_16x16x4_f32(false, a, false, b1, (short)0, acc1, false, false);
        acc2 = __builtin_amdgcn_wmma_f32_16x16x4_f32(false, a, false, b2, (short)0, acc2, false, false);
        acc3 = __builtin_amdgcn_wmma_f32_16x16x4_f32(false, a, false, b3, (short)0, acc3, false, false);
    }

    // Store 16x64 strip: row m = mt*16 + khalf*8 + i, cols {0,16,32,48}+row
    float* Cf = C + (size_t)f * N_ * FHID
                  + (size_t)(mt * 16 + khalf * 8) * FHID + row;
#pragma unroll
    for (int i = 0; i < 8; ++i) {
        Cf[i * FHID +  0] = acc0[i];
        Cf[i * FHID + 16] = acc1[i];
        Cf[i * FHID + 32] = acc2[i];
        Cf[i * FHID + 48] = acc3[i];
    }
}

// ---------------------------------------------------------------------------
// Scatter-add messages: acc[f][dst][c] += hW[f][src][c] * dis[src]*dis[dst]
// One (edge, feature) per thread. 64 features = threadIdx.x.
// ---------------------------------------------------------------------------
__global__ void scatter_kernel(const int* __restrict__ ei,
                               const float* __restrict__ dis,
                               const float* __restrict__ hW,
                               float* __restrict__ acc) {
    int c  = threadIdx.x;                                  // 0..63
    int eg = blockIdx.x * blockDim.y + threadIdx.y;        // f*E + e
    if (eg >= G_ * E_) return;
    int f = eg / E_, e = eg - f * E_;
    int s = ei[f * 2 * E_ + e];
    int d = ei[f * 2 * E_ + E_ + e];
    float norm = dis[f * N_ + s] * dis[f * N_ + d];
    atomicAdd(&acc[((size_t)f * N_ + d) * FHID + c],
              hW[((size_t)f * N_ + s) * FHID + c] * norm);
}

// conv1 epilogue: acc = relu(acc + hW*dis^2 (self-loop) + bias), in place
__global__ void relu_bias_kernel(const float* __restrict__ hW,
                                 const float* __restrict__ dis,
                                 const float* __restrict__ bias,
                                 float* __restrict__ acc) {
    size_t idx = (size_t)blockIdx.x * blockDim.x + threadIdx.x;
    if (idx >= (size_t)G_ * N_ * FHID) return;
    int    c  = (int)(idx & (FHID - 1));
    size_t fn = idx >> 6;                 // f*N_ + n
    float dn = dis[fn];
    float v  = acc[idx] + hW[idx] * dn * dn + bias[c];
    acc[idx] = v > 0.0f ? v : 0.0f;
}

// conv2 epilogue + mean pool: emb[f][c] = mean_n relu(acc + hW*dis^2 + b2)
// One block per (frame, feature); LDS tree reduction, direct store.
__global__ void relu_pool_kernel(const float* __restrict__ hW,
                                 const float* __restrict__ dis,
                                 const float* __restrict__ bias,
                                 const float* __restrict__ acc,
                                 float* __restrict__ emb) {
    int f = blockIdx.x, c = blockIdx.y, t = threadIdx.x;
    float bc = bias[c];
    float sum = 0.0f;
    for (int n = t; n < N_; n += 256) {
        size_t fn = (size_t)f * N_ + n;
        float dn = dis[fn];
        float v  = acc[fn * FHID + c] + hW[fn * FHID + c] * dn * dn + bc;
        sum += v > 0.0f ? v : 0.0f;
    }
    __shared__ float red[256];
    red[t] = sum;
    __syncthreads();
    for (int s = 128; s > 0; s >>= 1) {
        if (t < s) red[t] += red[t + s];
        __syncthreads();
    }
    if (t == 0) emb[f * FHID + c] = red[0] * (1.0f / (float)N_);
}

// ---------------------------------------------------------------------------
// GRU (torch gate order r,z,n) over T=30 steps + final FC. One block,
// 512 threads = (batch b = tid>>7, hidden j = tid&127). h kept in LDS.
// ---------------------------------------------------------------------------
__global__ void gru_fc_kernel(const float* __restrict__ emb,
                              const float* __restrict__ W_ih,
                              const float* __restrict__ W_hh,
                              const float* __restrict__ b_ih,
                              const float* __restrict__ b_hh,
                              const float* __restrict__ fc_w,
                              const float* __restrict__ fc_b,
                              float* __restrict__ out) {
    __shared__ float h[B_][FGRU];
    __shared__ float xt[B_][FHID];
    int tid = threadIdx.x;                 // 0..511
    int b = tid >> 7, j = tid & (FGRU - 1);
    h[b][j] = 0.0f;
    __syncthreads();

    for (int t = 0; t < T_; ++t) {
        if (tid < B_ * FHID) {
            int bb = tid / FHID, k = tid - bb * FHID;
            xt[bb][k] = emb[(bb * T_ + t) * FHID + k];   // frame f = b*T + t
        }
        __syncthreads();

        float gi_r = b_ih[j], gi_z = b_ih[j + FGRU], gi_n = b_ih[j + 2 * FGRU];
#pragma unroll 8
        for (int k = 0; k < FHID; ++k) {
            float x = xt[b][k];
            gi_r += x * W_ih[(j            ) * FHID + k];
            gi_z += x * W_ih[(j +     FGRU ) * FHID + k];
            gi_n += x * W_ih[(j + 2 * FGRU ) * FHID + k];
        }
        float gh_r = b_hh[j], gh_z = b_hh[j + FGRU], gh_n = b_hh[j + 2 * FGRU];
#pragma unroll 8
        for (int k = 0; k < FGRU; ++k) {
            float hv = h[b][k];
            gh_r += hv * W_hh[(j            ) * FGRU + k];
            gh_z += hv * W_hh[(j +     FGRU ) * FGRU + k];
            gh_n += hv * W_hh[(j + 2 * FGRU ) * FGRU + k];
        }
        float r  = 1.0f / (1.0f + expf(-(gi_r + gh_r)));
        float z  = 1.0f / (1.0f + expf(-(gi_z + gh_z)));
        float n  = tanhf(gi_n + r * gh_n);
        float hn = (1.0f - z) * n + z * h[b][j];
        __syncthreads();
        h[b][j] = hn;
        __syncthreads();
    }

    if (tid < B_ * NCLS) {
        int bb = tid >> 1, c = tid & 1;
        float s = fc_b[c];
        for (int k = 0; k < FGRU; ++k) s += h[bb][k] * fc_w[c * FGRU + k];
        out[bb * NCLS + c] = s;
    }
}

// ---------------------------------------------------------------------------
extern "C" void kernel_launch(void* const* d_in, const int* in_sizes, int n_in,
                              void* d_out, int out_size, void* d_ws, size_t ws_size,
                              hipStream_t stream) {
    const float* x    = (const float*)d_in[0];
    const int*   ei   = (const int*)  d_in[1];
    const float* W1   = (const float*)d_in[2];
    const float* b1   = (const float*)d_in[3];
    const float* W2   = (const float*)d_in[4];
    const float* b2   = (const float*)d_in[5];
    const float* W_ih = (const float*)d_in[6];
    const float* W_hh = (const float*)d_in[7];
    const float* b_ih = (const float*)d_in[8];
    const float* b_hh = (const float*)d_in[9];
    const float* fc_w = (const float*)d_in[10];
    const float* fc_b = (const float*)d_in[11];
    float* out = (float*)d_out;

    // Workspace layout (floats): dis | bufA | bufB | emb  (~124 MB)
    float* ws   = (float*)d_ws;
    float* dis  = ws;                                        // G_*N_
    float* bufA = dis  + (size_t)G_ * N_;                    // G_*N_*FHID
    float* bufB = bufA + (size_t)G_ * N_ * FHID;             // G_*N_*FHID
    float* emb  = bufB + (size_t)G_ * N_ * FHID;             // G_*FHID

    const size_t hbytes = sizeof(float) * (size_t)G_ * N_ * FHID;

    // degrees -> dis = rsqrt(deg+1)
    hipMemsetAsync(dis, 0, sizeof(float) * (size_t)G_ * N_, stream);
    deg_kernel<<<(G_ * E_ + 255) / 256, 256, 0, stream>>>(ei, dis);
    rsqrt_kernel<<<(G_ * N_ + 255) / 256, 256, 0, stream>>>(dis);

    const int mtiles = G_ * (N_ / 16);                       // 15000 strips
    const int gblk   = mtiles / 4;                           // 4 waves / block

    // ---- GCN conv 1 ----
    gemm_wmma<FIN><<<gblk, dim3(32, 4), 0, stream>>>(x, W1, bufA);
    hipMemsetAsync(bufB, 0, hbytes, stream);
    scatter_kernel<<<(G_ * E_ + 3) / 4, dim3(64, 4), 0, stream>>>(ei, dis, bufA, bufB);
    relu_bias_kernel<<<(int)(((size_t)G_ * N_ * FHID + 255) / 256), 256, 0, stream>>>(
        bufA, dis, b1, bufB);

    // ---- GCN conv 2 ----
    gemm_wmma<FHID><<<gblk, dim3(32, 4), 0, stream>>>(bufB, W2, bufA);
    hipMemsetAsync(bufB, 0, hbytes, stream);
    scatter_kernel<<<(G_ * E_ + 3) / 4, dim3(64, 4), 0, stream>>>(ei, dis, bufA, bufB);
    relu_pool_kernel<<<dim3(G_, FHID), 256, 0, stream>>>(bufA, dis, b2, bufB, emb);

    // ---- GRU + FC ----
    gru_fc_kernel<<<1, 512, 0, stream>>>(emb, W_ih, W_hh, b_ih, b_hh, fc_w, fc_b, out);
}